// Graph_Walker_19559281066069
// MI455X (gfx1250) — compile-verified
//
#include <hip/hip_runtime.h>
#include <math.h>

#define N_NODES 30460
#define D       64
#define B_GRP   128
#define K_CAND  6924
#define S_TOT   (B_GRP * K_CAND)
#define M_PER   50
#define A_HID   20
#define TILES   433               // ceil(K_CAND / 16)
#define TPW     8                 // tiles per wave (strip length)
#define STRIPS  55                // ceil(TILES / TPW)

typedef __attribute__((ext_vector_type(2))) float v2f;
typedef __attribute__((ext_vector_type(4))) float v4f;
typedef __attribute__((ext_vector_type(8))) float v8f;

// ---------------------------------------------------------------------------
// Kernel 1: user portrait (self-attention over 50 mentions per group)
// 128 blocks x 64 threads. Mentions staged in LDS.
// ---------------------------------------------------------------------------
__global__ void portrait_kernel(const float* __restrict__ graph,
                                const float* __restrict__ Wa,
                                const float* __restrict__ Va,
                                const int*   __restrict__ mention,
                                float*       __restrict__ portrait)
{
    __shared__ float me[M_PER * D];
    __shared__ float eb[M_PER];
    __shared__ float denom_s;
    const int b   = blockIdx.x;
    const int tid = threadIdx.x;          // 0..63

    for (int m = 0; m < M_PER; ++m) {
        int node = mention[b * M_PER + m];
        me[m * D + tid] = graph[node * D + tid];
    }
    __syncthreads();

    if (tid < M_PER) {
        float beta = 0.f;
        #pragma unroll 4
        for (int j = 0; j < A_HID; ++j) {
            float h = 0.f;
            for (int d = 0; d < D; ++d)
                h += me[tid * D + d] * Wa[d * A_HID + j];
            beta += tanhf(h) * Va[j];
        }
        eb[tid] = expf(beta);
    }
    __syncthreads();
    if (tid == 0) {
        float s = 0.f;
        for (int m = 0; m < M_PER; ++m) s += eb[m];
        denom_s = s;
    }
    __syncthreads();

    const float inv = 1.0f / denom_s;
    float acc = 0.f;
    for (int m = 0; m < M_PER; ++m)
        acc += eb[m] * inv * me[m * D + tid];
    portrait[b * D + tid] = acc;
}

// ---------------------------------------------------------------------------
// Kernel 2: per-group gate + query  q_b = w*utter + (1-w)*portrait
// ---------------------------------------------------------------------------
__global__ void context_kernel(const float* __restrict__ graph,
                               const float* __restrict__ abstract_row,
                               const float* __restrict__ utter,
                               const float* __restrict__ portrait,
                               const float* __restrict__ W1w, const float* __restrict__ W1b,
                               const float* __restrict__ W2w, const float* __restrict__ W2b,
                               const int* __restrict__ last0, const int* __restrict__ intent0,
                               const int* __restrict__ last1, const int* __restrict__ intent1,
                               float* __restrict__ q)   // [2][B][D]
{
    __shared__ float red[D];
    __shared__ float w_s;
    const int b   = blockIdx.x;
    const int tid = threadIdx.x;          // 0..63

    const float u = utter[b * D + tid];
    const float p = portrait[b * D + tid];

    for (int l = 0; l < 2; ++l) {
        const float* Ww = l ? W2w : W1w;
        const float* Wb = l ? W2b : W1b;
        const int li = l ? last1[b] : last0[b];
        const int it = l ? intent1[b] : intent0[b];
        const float* srow = (li < N_NODES) ? (graph + (long)li * D) : abstract_row;
        const float s = srow[tid];

        red[tid] = u * Ww[it * 192 + tid]
                 + p * Ww[it * 192 + 64 + tid]
                 + s * Ww[it * 192 + 128 + tid];
        __syncthreads();
        if (tid == 0) {
            float dot = 0.f;
            for (int d = 0; d < D; ++d) dot += red[d];
            w_s = 1.0f / (1.0f + expf(-(dot + Wb[it])));
        }
        __syncthreads();
        const float w = w_s;
        q[(l * B_GRP + b) * D + tid] = u * w + p * (1.0f - w);
        __syncthreads();
    }
}

// ---------------------------------------------------------------------------
// Kernel 3: scores. One wave per strip of TPW 16-candidate tiles of one
// (layer, group). The 16 query B-fragments are loaded ONCE per wave and held
// in registers (32 VGPRs); per tile only the 16 gathered A-row fragments hit
// memory. All TPW row indices are preloaded so the per-tile critical path is
// gather -> v_wmma_f32_16x16x4_f32 (x16, chained accumulator).
// A layout (ISA 7.12.2, 32-bit A 16x4): lanes 0-15 -> K=k0,k0+1;
// lanes 16-31 -> K=k0+2,k0+3. B mirrored (columns identical -> N-map moot).
// ---------------------------------------------------------------------------
__global__ void score_kernel(const float* __restrict__ graph,
                             const float* __restrict__ abstract_row,
                             const float* __restrict__ q,      // [2][B][D]
                             const int*   __restrict__ sel0,
                             const int*   __restrict__ sel1,
                             float*       __restrict__ out)    // [2][S]
{
    const int wg   = blockIdx.x * (blockDim.x >> 5) + (threadIdx.x >> 5);
    const int lane = threadIdx.x & 31;
    const int total = 2 * B_GRP * STRIPS;
    if (wg >= total) return;              // wave-uniform

    const int layer = wg / (B_GRP * STRIPS);
    const int rem   = wg - layer * (B_GRP * STRIPS);
    const int b     = rem / STRIPS;
    const int strip = rem - b * STRIPS;
    const int tile0 = strip * TPW;

    const int* selb = (layer ? sel1 : sel0) + b * K_CAND;
    const int row  = lane & 15;
    const int half = lane >> 4;
    const int ko   = half * 2;
    const float* qrow = q + (layer * B_GRP + b) * D;

    // --- query fragments: loaded once, register-resident for all TPW tiles
    v2f bv[16];
    #pragma unroll
    for (int c = 0; c < 16; ++c)
        bv[c] = *(const v2f*)(qrow + 4 * c + ko);

    // --- preload this lane's row index for every tile in the strip
    int node[TPW];
    #pragma unroll
    for (int t = 0; t < TPW; ++t) {
        int cand = (tile0 + t) * 16 + row;
        if (cand >= K_CAND) cand = K_CAND - 1;   // clamp (dup row, masked at store)
        node[t] = selb[cand];
    }

    for (int t = 0; t < TPW; ++t) {
        const int tile = tile0 + t;
        if (tile >= TILES) break;                // wave-uniform tail

        const float* arow = (node[t] < N_NODES) ? (graph + (long)node[t] * D)
                                                : abstract_row;
        v8f acc = {};
        #pragma unroll
        for (int c = 0; c < 16; ++c) {
            v2f av = *(const v2f*)(arow + 4 * c + ko);
            acc = __builtin_amdgcn_wmma_f32_16x16x4_f32(
                      false, av, false, bv[c], (short)0, acc, false, false);
        }

        // D layout: VGPR v, lanes 0-15 -> M=v; lanes 16-31 -> M=v+8 (cols equal)
        if ((lane & 15) == 0) {                  // lane 0: cands 0-7, lane 16: 8-15
            const int off = half * 8;
            float* o = out + (long)layer * S_TOT + (long)b * K_CAND
                           + tile * 16 + off;
            if (tile * 16 + 16 <= K_CAND) {      // full tile: 2x b128 stores
                v4f lo = __builtin_shufflevector(acc, acc, 0, 1, 2, 3);
                v4f hi = __builtin_shufflevector(acc, acc, 4, 5, 6, 7);
                *(v4f*)(o)     = lo;
                *(v4f*)(o + 4) = hi;
            } else {                             // ragged last tile of the group
                #pragma unroll
                for (int i = 0; i < 8; ++i)
                    if (tile * 16 + off + i < K_CAND) o[i] = acc[i];
            }
        }
    }
}

// ---------------------------------------------------------------------------
extern "C" void kernel_launch(void* const* d_in, const int* in_sizes, int n_in,
                              void* d_out, int out_size, void* d_ws, size_t ws_size,
                              hipStream_t stream)
{
    const float* graph   = (const float*)d_in[0];
    const float* utter   = (const float*)d_in[1];
    const float* abstr   = (const float*)d_in[2];
    const float* Wa      = (const float*)d_in[3];
    const float* Va      = (const float*)d_in[4];
    const float* W1w     = (const float*)d_in[5];
    const float* W1b     = (const float*)d_in[6];
    const float* W2w     = (const float*)d_in[7];
    const float* W2b     = (const float*)d_in[8];
    const int*   mention = (const int*)d_in[9];
    // d_in[10] mention_batch_index: implicit (repeat(arange(B), 50))
    const int*   sel0    = (const int*)d_in[11];
    const int*   last0   = (const int*)d_in[15];
    const int*   intent0 = (const int*)d_in[16];
    const int*   sel1    = (const int*)d_in[17];
    const int*   last1   = (const int*)d_in[21];
    const int*   intent1 = (const int*)d_in[22];
    float* out = (float*)d_out;

    float* portrait = (float*)d_ws;                 // 128*64 f32
    float* q        = portrait + B_GRP * D;         // 2*128*64 f32  (96 KB total)

    portrait_kernel<<<B_GRP, 64, 0, stream>>>(graph, Wa, Va, mention, portrait);
    context_kernel<<<B_GRP, 64, 0, stream>>>(graph, abstr, utter, portrait,
                                             W1w, W1b, W2w, W2b,
                                             last0, intent0, last1, intent1, q);

    const int total_waves = 2 * B_GRP * STRIPS;     // 14,080
    const int wpb = 8;                              // 256 threads / block
    const int blocks = (total_waves + wpb - 1) / wpb;
    score_kernel<<<blocks, wpb * 32, 0, stream>>>(graph, abstr, q, sel0, sel1, out);
}